// GraphConv_81621558493743
// MI455X (gfx1250) — compile-verified
//
#include <hip/hip_runtime.h>
#include <hip/hip_bf16.h>

// Problem constants (match the JAX reference)
#define BB 8
#define NN 512
#define PP 32
#define LL 3

typedef __attribute__((ext_vector_type(2))) float v2f;
typedef __attribute__((ext_vector_type(8))) float v8f;

// ---------------------------------------------------------------------------
// Kernel 1: edge term, one pass over weights for ALL layers.
// t31[i][b][p][n] = sum_m relu( t4[i][p][0]*x[b,n] + t4[i][p][1]*w[b,n,m]
//                             + t4[i][p][2]*(0.5 - 0.5*s_n*s_m) + t4[i][p][3] )
// One block per (b,n) row: 8 waves, wave g strides m, lane = p.
// ---------------------------------------------------------------------------
__global__ void edge_kernel(const float* __restrict__ x,
                            const float* __restrict__ w,
                            const float* __restrict__ t4,
                            float* __restrict__ t31) {
    const int row = blockIdx.x;          // b*NN + n
    const int b   = row >> 9;
    const int n   = row & (NN - 1);
    const int p   = threadIdx.x & 31;    // lane = p (wave32)
    const int g   = threadIdx.x >> 5;    // wave id 0..7

    __shared__ float ssm[NN];            // s_m = 2*x[b,m]-1
    for (int m = threadIdx.x; m < NN; m += 256)
        ssm[m] = 2.0f * x[b * NN + m] - 1.0f;
    __syncthreads();

    const float xv = x[b * NN + n];
    const float sn = 2.0f * xv - 1.0f;

    float Aw[LL], Bs[LL], Cc[LL];
#pragma unroll
    for (int i = 0; i < LL; ++i) {
        const float* t = t4 + (i * PP + p) * 4;
        const float t0 = t[0], tw = t[1], tc = t[2], tb = t[3];
        Aw[i] = tw;                      // coeff of w_nm
        Bs[i] = -0.5f * tc * sn;         // coeff of s_m
        Cc[i] = fmaf(t0, xv, tb + 0.5f * tc);
    }

    float acc[LL] = {0.f, 0.f, 0.f};
    const float* wrow = w + (size_t)row * NN;
    for (int m = g; m < NN; m += 8) {
        const float wm = wrow[m];        // broadcast across lanes
        const float sm = ssm[m];
#pragma unroll
        for (int i = 0; i < LL; ++i) {
            const float arg = fmaf(Aw[i], wm, fmaf(Bs[i], sm, Cc[i]));
            acc[i] += fmaxf(arg, 0.0f);
        }
    }

    __shared__ float red[8][LL][PP];
#pragma unroll
    for (int i = 0; i < LL; ++i) red[g][i][p] = acc[i];
    __syncthreads();

    if (g < LL) {                        // waves 0..2 finalize layer i=g
        const int i = g;
        float s = 0.f;
#pragma unroll
        for (int gg = 0; gg < 8; ++gg) s += red[gg][i][p];
        t31[(((size_t)i * BB + b) * PP + p) * NN + n] = s;
    }
}

// ---------------------------------------------------------------------------
// Kernel 2: small WMMA GEMM, full-M per wave (32x16 output block, 2 accs).
//   out[l][b][p][n] = (opt bias_l) + sum_q W_l[p,q] * f(V_l[b,q,n])
//   W: (layers,32,32), V: (layers,B,32,N).  f = relu if reluV else identity.
//   If t1 != null adds term1[b,p,n] = t1_l[p,0]*(1-x) + t1_l[p,1] + t1_l[p,2]*extra.
// Layer = blockIdx.y. Tiles per layer = B*32 = 256; 8 waves/block -> 32 blocks.
// ---------------------------------------------------------------------------
__global__ void smallgemm_kernel(const float* __restrict__ W0, int wStride,
                                 const float* __restrict__ V0, size_t vStride,
                                 float* __restrict__ out0, size_t oStride,
                                 const float* __restrict__ t10,
                                 const float* __restrict__ x,
                                 const float* __restrict__ extra,
                                 int reluV) {
    const int layer = blockIdx.y;
    const float* W   = W0 + (size_t)layer * wStride;
    const float* V   = V0 + (size_t)layer * vStride;
    float*       out = out0 + (size_t)layer * oStride;
    const float* t1  = t10 ? (t10 + (size_t)layer * PP * 3) : nullptr;

    const int lane = threadIdx.x & 31;
    const int l16  = lane & 15;
    const int hi   = lane >> 4;                 // 0 or 1
    const int tile = blockIdx.x * 8 + (threadIdx.x >> 5);   // 0..255
    const int b    = tile >> 5;
    const int nbase = (tile & 31) * 16;
    const int coln = nbase + l16;

    const float* Vb = V + (size_t)b * PP * NN;

    v8f acc0 = {}, acc1 = {};
#pragma unroll
    for (int k = 0; k < PP; k += 4) {
        const int ka = k + 2 * hi;
        v2f a0 = *(const v2f*)(W + l16 * PP + ka);          // rows 0..15
        v2f a1 = *(const v2f*)(W + (16 + l16) * PP + ka);   // rows 16..31
        v2f bm;
        bm[0] = Vb[(size_t)ka * NN + coln];
        bm[1] = Vb[(size_t)(ka + 1) * NN + coln];
        if (reluV) { bm[0] = fmaxf(bm[0], 0.f); bm[1] = fmaxf(bm[1], 0.f); }
        acc0 = __builtin_amdgcn_wmma_f32_16x16x4_f32(
            false, a0, false, bm, (short)0, acc0, false, false);
        acc1 = __builtin_amdgcn_wmma_f32_16x16x4_f32(
            false, a1, false, bm, (short)0, acc1, false, false);
    }

    float xv = 0.f, ev = 0.f;
    if (t1) { xv = x[b * NN + coln]; ev = extra[b * NN + coln]; }
#pragma unroll
    for (int r = 0; r < 8; ++r) {
        const int r0 = r + 8 * hi;            // rows of acc0
        const int r1 = 16 + r0;               // rows of acc1
        float v0 = acc0[r], v1 = acc1[r];
        if (t1) {
            const float* tp0 = t1 + r0 * 3;
            const float* tp1 = t1 + r1 * 3;
            v0 += fmaf(tp0[0], 1.0f - xv, fmaf(tp0[2], ev, tp0[1]));
            v1 += fmaf(tp1[0], 1.0f - xv, fmaf(tp1[2], ev, tp1[1]));
        }
        out[((size_t)b * PP + r0) * NN + coln] = v0;
        out[((size_t)b * PP + r1) * NN + coln] = v1;
    }
}

// ---------------------------------------------------------------------------
// Kernel 3: big WMMA GEMM, full-M per wave (32x16 output block, 2 accs).
//   mu_out[b,p,m] = relu(Cbias[b,p,m] + sum_n X[b,p,n]*A[b,n,m])
//   X: (B,32,512), Adj: (B,512,512). K=512. Each adjacency element is read
//   exactly once (shared B-frag across both M-halves). Tiles = B*32 = 256.
// ---------------------------------------------------------------------------
__global__ void bigterm_kernel(const float* __restrict__ X,
                               const float* __restrict__ Adj,
                               const float* __restrict__ Cbias,
                               float* __restrict__ mu_out) {
    const int lane = threadIdx.x & 31;
    const int l16  = lane & 15;
    const int hi   = lane >> 4;
    const int tile = blockIdx.x * 8 + (threadIdx.x >> 5);   // 0..255
    const int b    = tile >> 5;
    const int nbase = (tile & 31) * 16;
    const int coln = nbase + l16;

    const float* Xb = X   + (size_t)b * PP * NN;
    const float* Ab = Adj + (size_t)b * NN * NN;

    v8f acc0 = {}, acc1 = {};
#pragma unroll 4
    for (int k = 0; k < NN; k += 4) {
        const int ka = k + 2 * hi;
        v2f a0 = *(const v2f*)(Xb + (size_t)l16 * NN + ka);        // rows 0..15
        v2f a1 = *(const v2f*)(Xb + (size_t)(16 + l16) * NN + ka); // rows 16..31
        v2f bm;
        bm[0] = Ab[(size_t)ka * NN + coln];
        bm[1] = Ab[(size_t)(ka + 1) * NN + coln];
        // warm L2/L1 ahead (clamped in-bounds); emits global_prefetch_b8
        const int kp = (k + 64 < NN) ? (k + 64) : (NN - 4);
        __builtin_prefetch(Ab + (size_t)(kp + 2 * hi) * NN + coln, 0, 1);
        acc0 = __builtin_amdgcn_wmma_f32_16x16x4_f32(
            false, a0, false, bm, (short)0, acc0, false, false);
        acc1 = __builtin_amdgcn_wmma_f32_16x16x4_f32(
            false, a1, false, bm, (short)0, acc1, false, false);
    }

#pragma unroll
    for (int r = 0; r < 8; ++r) {
        const int r0 = r + 8 * hi;
        const int r1 = 16 + r0;
        const size_t off0 = ((size_t)b * PP + r0) * NN + coln;
        const size_t off1 = ((size_t)b * PP + r1) * NN + coln;
        mu_out[off0] = fmaxf(acc0[r] + Cbias[off0], 0.0f);
        mu_out[off1] = fmaxf(acc1[r] + Cbias[off1], 0.0f);
    }
}

// ---------------------------------------------------------------------------
extern "C" void kernel_launch(void* const* d_in, const int* in_sizes, int n_in,
                              void* d_out, int out_size, void* d_ws, size_t ws_size,
                              hipStream_t stream) {
    const float* x     = (const float*)d_in[0];   // (B,N)
    const float* adj   = (const float*)d_in[1];   // (B,N,N)
    const float* wts   = (const float*)d_in[2];   // (B,N,N)
    const float* extra = (const float*)d_in[3];   // (B,N,1)
    const float* t1    = (const float*)d_in[4];   // (L,P,3)
    const float* t2    = (const float*)d_in[5];   // (L,P,P)
    const float* t3    = (const float*)d_in[6];   // (L,P,P)
    const float* t4    = (const float*)d_in[7];   // (L,P,4)
    float* out = (float*)d_out;                   // (B,P,N)

    const size_t perLayer = (size_t)BB * PP * NN;       // 131072
    float* t31   = (float*)d_ws;                        // L*B*P*N
    float* cbias = t31 + LL * perLayer;                 // L*B*P*N
    float* mu    = cbias + LL * perLayer;               // B*P*N
    float* Xbuf  = mu + perLayer;                       // B*P*N

    // 1) edge term for all layers (single pass over weights)
    edge_kernel<<<BB * NN, 256, 0, stream>>>(x, wts, t4, t31);

    // 2) Cbias[i] = term1_i + t3_i @ t31_i  -- ONE launch, gridDim.y = L
    smallgemm_kernel<<<dim3(32, LL), 256, 0, stream>>>(
        t3, PP * PP, t31, perLayer, cbias, perLayer, t1, x, extra, /*reluV=*/0);

    // 3) layer 1: X = t2_1 @ relu(Cbias_0)   (layer-0 relu fused into V load)
    smallgemm_kernel<<<dim3(32, 1), 256, 0, stream>>>(
        t2 + 1 * PP * PP, 0, cbias, 0, Xbuf, 0, nullptr, nullptr, nullptr, /*reluV=*/1);
    //    mu_1 = relu(Cbias_1 + X @ Adj)
    bigterm_kernel<<<32, 256, 0, stream>>>(Xbuf, adj, cbias + perLayer, mu);

    // 4) layer 2: X = t2_2 @ mu_1 ; out = relu(Cbias_2 + X @ Adj)
    smallgemm_kernel<<<dim3(32, 1), 256, 0, stream>>>(
        t2 + 2 * PP * PP, 0, mu, 0, Xbuf, 0, nullptr, nullptr, nullptr, /*reluV=*/0);
    bigterm_kernel<<<32, 256, 0, stream>>>(Xbuf, adj, cbias + 2 * perLayer, out);
}